// GNNModel_41326175322384
// MI455X (gfx1250) — compile-verified
//
#include <hip/hip_runtime.h>

// ---------------------------------------------------------------------------
// 3-layer GCN for MI455X (gfx1250).
//   Memory-bound: ~4 GB of gather/scatter traffic vs 4.1 GFLOP of GEMM.
//   GEMMs use V_WMMA_F32_16X16X4_F32 (exact fp32 matrix pipe, wave32).
//   Scatter-add uses coalesced GLOBAL_ATOMIC_ADD_F32.
// ---------------------------------------------------------------------------

typedef __attribute__((ext_vector_type(2))) float v2f;
typedef __attribute__((ext_vector_type(8))) float v8f;

// ---------------- degree / normalization ----------------

__global__ __launch_bounds__(256) void init_deg_kernel(float* deg, int n) {
    int i = blockIdx.x * 256 + threadIdx.x;
    if (i < n) deg[i] = 1.0f;  // self-loop contributes 1 to every degree
}

__global__ __launch_bounds__(256) void edge_deg_kernel(const int* __restrict__ dst,
                                                       float* deg, int e) {
    int i = blockIdx.x * 256 + threadIdx.x;
    if (i < e) atomicAdd(&deg[dst[i]], 1.0f);
}

__global__ __launch_bounds__(256) void rsqrt_kernel(float* deg, int n) {
    int i = blockIdx.x * 256 + threadIdx.x;
    if (i < n) deg[i] = rsqrtf(deg[i]);  // deg >= 1 always, no zero guard needed
}

// ---------------- WMMA fp32 GEMM: C[16 x NT*16] per wave ----------------
// One wave (32 threads) owns a 16-row stripe and the full output width.
// A-frag layout (16x4 f32, ISA 7.12.2): lane<16 -> K=k0,k0+1 ; lane>=16 -> K=k0+2,k0+3
// B-frag layout (4x16 f32): VGPR0 = rows k0 / k0+2 (lane halves), VGPR1 = k0+1 / k0+3
// D layout: VGPR v -> row v (lanes 0-15) / row v+8 (lanes 16-31)
template <int NT>
__global__ __launch_bounds__(32) void gemm_wmma_f32(const float* __restrict__ A,
                                                    const float* __restrict__ W,
                                                    float* __restrict__ C,
                                                    int K, int reluA) {
    const int Nout = NT * 16;
    const int tileM = blockIdx.x;
    const int lane  = threadIdx.x;
    const int half  = lane >> 4;   // 0 or 1
    const int l16   = lane & 15;

    const float* Arow = A + ((size_t)tileM * 16 + l16) * (size_t)K;

    v8f acc[NT];
#pragma unroll
    for (int t = 0; t < NT; ++t) acc[t] = (v8f){};

    for (int k0 = 0; k0 < K; k0 += 4) {
        const int ka = k0 + 2 * half;
        float a0 = Arow[ka];
        float a1 = Arow[ka + 1];
        if (reluA) { a0 = fmaxf(a0, 0.0f); a1 = fmaxf(a1, 0.0f); }  // fused ReLU of prev layer
        v2f a = {a0, a1};
        const float* Wk0 = W + (size_t)ka * Nout + l16;
        const float* Wk1 = Wk0 + Nout;
#pragma unroll
        for (int t = 0; t < NT; ++t) {
            v2f b = {Wk0[t * 16], Wk1[t * 16]};
            // (neg_a, A, neg_b, B, c_mod, C, reuse_a, reuse_b)
            acc[t] = __builtin_amdgcn_wmma_f32_16x16x4_f32(
                false, a, false, b, (short)0, acc[t], false, false);
        }
    }

#pragma unroll
    for (int t = 0; t < NT; ++t) {
        float* Cp = C + ((size_t)tileM * 16 + half * 8) * Nout + t * 16 + l16;
#pragma unroll
        for (int v = 0; v < 8; ++v) Cp[(size_t)v * Nout] = acc[t][v];
    }
}

// ---------------- self-loop init: out = bias + h * dinv^2 ----------------
__global__ __launch_bounds__(256) void self_init_kernel(const float* __restrict__ h,
                                                        const float* __restrict__ dinv,
                                                        const float* __restrict__ bias,
                                                        float* __restrict__ out,
                                                        int n, int d) {
    const int d4 = d >> 2;  // d is 128 or 64, both multiples of 4
    long long idx = (long long)blockIdx.x * 256 + threadIdx.x;
    long long total = (long long)n * d4;
    if (idx >= total) return;
    int node = (int)(idx / d4);
    int c4   = (int)(idx % d4);
    float w = dinv[node];
    w = w * w;
    float4 hv = ((const float4*)(h + (size_t)node * d))[c4];
    float4 bv = ((const float4*)bias)[c4];
    float4 o  = make_float4(bv.x + hv.x * w, bv.y + hv.y * w,
                            bv.z + hv.z * w, bv.w + hv.w * w);
    ((float4*)(out + (size_t)node * d))[c4] = o;
}

// ---------------- edge scatter: out[dst] += h[src] * dinv[s]*dinv[d] ----------------
// One wave32 per edge; lanes cover channels c = lane + 32*j (coalesced per atomic op).
__global__ __launch_bounds__(256) void scatter_kernel(const float* __restrict__ h,
                                                      const float* __restrict__ dinv,
                                                      const int* __restrict__ src,
                                                      const int* __restrict__ dst,
                                                      float* out, int e, int d) {
    const int wave = threadIdx.x >> 5;
    const int lane = threadIdx.x & 31;
    long long edge = (long long)blockIdx.x * 8 + wave;
    if (edge >= e) return;
    const int s = src[edge];
    const int t = dst[edge];
    const float norm = dinv[s] * dinv[t];
    const float* hs = h + (size_t)s * d;
    float* od = out + (size_t)t * d;
    for (int c = lane; c < d; c += 32) {
        atomicAdd(&od[c], hs[c] * norm);
    }
}

// ---------------------------------------------------------------------------

extern "C" void kernel_launch(void* const* d_in, const int* in_sizes, int n_in,
                              void* d_out, int out_size, void* d_ws, size_t ws_size,
                              hipStream_t stream) {
    const float* x  = (const float*)d_in[0];
    const int*   ei = (const int*)d_in[1];
    const float* W1 = (const float*)d_in[2];
    const float* b1 = (const float*)d_in[3];
    const float* W2 = (const float*)d_in[4];
    const float* b2 = (const float*)d_in[5];
    const float* W3 = (const float*)d_in[6];
    const float* b3 = (const float*)d_in[7];
    float* out = (float*)d_out;

    const int N = in_sizes[0] / 128;  // 50000 (multiple of 16)
    const int E = in_sizes[1] / 2;    // 1600000
    const int* src = ei;
    const int* dst = ei + E;

    // Workspace: dinv (padded to 50176 floats for alignment), h, agg
    float* dinv = (float*)d_ws;
    float* h    = dinv + 50176;
    float* agg  = h + (size_t)N * 128;

    const int TB = 256;
    const int nbN = (N + TB - 1) / TB;
    const int nbE = (E + TB - 1) / TB;
    const int nbEw = (E + 7) / 8;           // 8 waves per block, 1 edge per wave

    // --- degrees + symmetric normalization (graph is fixed across layers) ---
    init_deg_kernel<<<nbN, TB, 0, stream>>>(dinv, N);
    edge_deg_kernel<<<nbE, TB, 0, stream>>>(dst, dinv, E);
    rsqrt_kernel<<<nbN, TB, 0, stream>>>(dinv, N);

    const int nbSelf128 = (int)(((long long)N * 32 + TB - 1) / TB);  // N*128/4 elems
    const int nbSelf64  = (int)(((long long)N * 16 + TB - 1) / TB);  // N*64/4 elems

    // --- layer 1: h = x @ W1 ; agg = b1 + norm-weighted aggregation ---
    gemm_wmma_f32<8><<<dim3(N / 16), 32, 0, stream>>>(x, W1, h, 128, 0);
    self_init_kernel<<<nbSelf128, TB, 0, stream>>>(h, dinv, b1, agg, N, 128);
    scatter_kernel<<<nbEw, TB, 0, stream>>>(h, dinv, src, dst, agg, E, 128);

    // --- layer 2: h = relu(agg) @ W2 (ReLU fused into GEMM A-load) ---
    gemm_wmma_f32<8><<<dim3(N / 16), 32, 0, stream>>>(agg, W2, h, 128, 1);
    self_init_kernel<<<nbSelf128, TB, 0, stream>>>(h, dinv, b2, agg, N, 128);
    scatter_kernel<<<nbEw, TB, 0, stream>>>(h, dinv, src, dst, agg, E, 128);

    // --- layer 3: h = relu(agg) @ W3, aggregate straight into d_out ---
    gemm_wmma_f32<4><<<dim3(N / 16), 32, 0, stream>>>(agg, W3, h, 128, 1);
    self_init_kernel<<<nbSelf64, TB, 0, stream>>>(h, dinv, b3, out, N, 64);
    scatter_kernel<<<nbEw, TB, 0, stream>>>(h, dinv, src, dst, out, E, 64);
}